// Detect_22265110462825
// MI455X (gfx1250) — compile-verified
//
#include <hip/hip_runtime.h>
#include <hip/hip_bf16.h>

typedef __attribute__((ext_vector_type(2))) float v2f;
typedef __attribute__((ext_vector_type(8))) float v8f;

#define NC_      80
#define A_TOT    8400
#define BATCH    8
#define MAX_DET_ 300

__device__ __forceinline__ float silu_f(float v) { return v / (1.0f + __expf(-v)); }

// ---------------------------------------------------------------------------
// Implicit-GEMM conv (1x1 or 3x3 'SAME') using V_WMMA_F32_16X16X4_F32.
// Block: 256 threads (8 waves). Block tile: M=64 (Cout) x N=64 (pixels).
// Wave grid 4(M) x 2(N): each wave owns a 16x32 strip -> two v8f accumulators
// sharing one A fragment per K-step. K staged through LDS in chunks of 32
// (all K dims in this net are multiples of 32), 16 WMMAs per wave per chunk.
//
// f32 WMMA lane layout (ISA 7.12.2):
//   A (16x4):  lane<16: {A[m][k0],A[m][k0+1]}  lane>=16: {A[m][k0+2],A[m][k0+3]}
//   B (4x16):  lane<16: {B[k0][n],B[k0+1][n]}  lane>=16: {B[k0+2][n],B[k0+3][n]}
//   C/D:       vgpr j, lane L: row = j + (L>=16 ? 8 : 0), col = L&15
// ---------------------------------------------------------------------------
__global__ __launch_bounds__(256) void conv_wmma(
    const float* __restrict__ in,    // (B, Cin, H, W)
    const float* __restrict__ wgt,   // (Cout, Cin*KK) flattened OIHW
    const float* __restrict__ bias,  // (Cout)
    float* __restrict__ out,         // (B, Cout, outA) at aOff
    int Cin, int Cout, int H, int W, int k3, int dosilu, int outA, int aOff)
{
  const int HW   = H * W;
  const int Ktot = Cin * (k3 ? 9 : 1);
  const int b       = blockIdx.z;
  const int coBase  = blockIdx.y * 64;
  const int pixBase = blockIdx.x * 64;
  const int tid  = threadIdx.x;
  const int lane = tid & 31;
  const int wv   = tid >> 5;
  const int moff  = (wv >> 1) * 16;   // 0,16,32,48
  const int noff0 = (wv & 1) * 32;    // 0,32  (wave's 32-wide pixel strip)

  __shared__ float sA[64][33];  // [m][k], padded against bank conflicts
  __shared__ float sB[32][65];  // [k][n]

  v8f acc0 = {0.f,0.f,0.f,0.f,0.f,0.f,0.f,0.f};
  v8f acc1 = {0.f,0.f,0.f,0.f,0.f,0.f,0.f,0.f};

  const int l     = lane & 15;
  const int khalf = (lane >> 4) << 1;   // 0 or 2

  for (int k0 = 0; k0 < Ktot; k0 += 32) {
    // stage 64x32 weight tile (zero-pad rows beyond Cout)
    for (int i = tid; i < 64 * 32; i += 256) {
      int m = i >> 5, kk = i & 31;
      int co = coBase + m;
      sA[m][kk] = (co < Cout) ? wgt[(size_t)co * Ktot + (k0 + kk)] : 0.f;
    }
    // prefetch next weight chunk (global_prefetch_b8)
    if (k0 + 32 < Ktot && tid < 64 && (coBase + tid) < Cout)
      __builtin_prefetch(&wgt[(size_t)(coBase + tid) * Ktot + k0 + 32], 0, 1);
    // stage 32x64 input tile (im2col gather for 3x3)
    for (int i = tid; i < 32 * 64; i += 256) {
      int kk = i >> 6, n = i & 63;
      int gk = k0 + kk;
      int p  = pixBase + n;
      float v = 0.f;
      if (p < HW) {
        if (k3) {
          int ci = gk / 9;
          int r  = gk - ci * 9;
          int ky = r / 3, kx = r - ky * 3;
          int py = p / W, px = p - py * W;
          int iy = py + ky - 1, ix = px + kx - 1;
          if (iy >= 0 && iy < H && ix >= 0 && ix < W)
            v = in[((size_t)(b * Cin + ci) * H + iy) * W + ix];
        } else {
          v = in[(size_t)(b * Cin + gk) * HW + p];
        }
      }
      sB[kk][n] = v;
    }
    __syncthreads();
#pragma unroll
    for (int ks = 0; ks < 32; ks += 4) {
      v2f a, bf0, bf1;
      a.x   = sA[moff + l][ks + khalf];
      a.y   = sA[moff + l][ks + khalf + 1];
      bf0.x = sB[ks + khalf][noff0 + l];
      bf0.y = sB[ks + khalf + 1][noff0 + l];
      bf1.x = sB[ks + khalf][noff0 + 16 + l];
      bf1.y = sB[ks + khalf + 1][noff0 + 16 + l];
      acc0 = __builtin_amdgcn_wmma_f32_16x16x4_f32(
          false, a, false, bf0, (short)0, acc0, false, false);
      acc1 = __builtin_amdgcn_wmma_f32_16x16x4_f32(
          false, a, false, bf1, (short)0, acc1, false, false);
    }
    __syncthreads();
  }

  const int rowsel = (lane >> 4) << 3;  // 0 or 8
  const int ncol0  = pixBase + noff0 + l;
  const int ncol1  = ncol0 + 16;
#pragma unroll
  for (int j = 0; j < 8; ++j) {
    int co = coBase + moff + rowsel + j;
    if (co < Cout) {
      float bsv = bias[co];
      size_t rowBase = ((size_t)b * Cout + co) * outA + aOff;
      if (ncol0 < HW) {
        float v = acc0[j] + bsv;
        if (dosilu) v = silu_f(v);
        out[rowBase + ncol0] = v;
      }
      if (ncol1 < HW) {
        float v = acc1[j] + bsv;
        if (dosilu) v = silu_f(v);
        out[rowBase + ncol1] = v;
      }
    }
  }
}

// ---------------------------------------------------------------------------
// Depthwise 3x3 'SAME' conv + SiLU (memory-bound -> plain VALU)
// ---------------------------------------------------------------------------
__global__ void dwconv3x3_silu(const float* __restrict__ in,
                               const float* __restrict__ wgt,
                               const float* __restrict__ bias,
                               float* __restrict__ out,
                               int C, int H, int W, int total)
{
  int t = blockIdx.x * blockDim.x + threadIdx.x;
  if (t >= total) return;
  int x = t % W;
  int y = (t / W) % H;
  int plane = t / (W * H);       // b*C + c
  int c = plane % C;
  const float* wp = wgt + c * 9;
  const float* ip = in + (size_t)plane * H * W;
  float acc = bias[c];
#pragma unroll
  for (int ky = 0; ky < 3; ++ky) {
    int iy = y + ky - 1;
    if (iy < 0 || iy >= H) continue;
#pragma unroll
    for (int kx = 0; kx < 3; ++kx) {
      int ix = x + kx - 1;
      if (ix < 0 || ix >= W) continue;
      acc += wp[ky * 3 + kx] * ip[iy * W + ix];
    }
  }
  out[t] = silu_f(acc);
}

// ---------------------------------------------------------------------------
// DFL softmax decode + sigmoid + per-anchor max score.
// boxAll (B,64,8400), scoreAll (B,80,8400) -> ybuf (B,8400,84), maxsc (B,8400)
// ---------------------------------------------------------------------------
__global__ void decode_kernel(const float* __restrict__ boxAll,
                              const float* __restrict__ scoreAll,
                              float* __restrict__ ybuf,
                              float* __restrict__ maxsc)
{
  int t = blockIdx.x * blockDim.x + threadIdx.x;
  if (t >= BATCH * A_TOT) return;
  int b = t / A_TOT;
  int a = t - b * A_TOT;

  float s; int w, loc;
  if (a < 6400)      { s = 8.f;  w = 80; loc = a; }
  else if (a < 8000) { s = 16.f; w = 40; loc = a - 6400; }
  else               { s = 32.f; w = 20; loc = a - 8000; }
  float ay = (float)(loc / w) + 0.5f;
  float ax = (float)(loc % w) + 0.5f;

  const float* bb = boxAll + (size_t)b * 64 * A_TOT + a;
  float dist[4];
#pragma unroll
  for (int f = 0; f < 4; ++f) {
    float vals[16];
    float m = -__builtin_inff();
#pragma unroll
    for (int r = 0; r < 16; ++r) {
      float v = bb[(size_t)(f * 16 + r) * A_TOT];
      vals[r] = v;
      m = fmaxf(m, v);
    }
    float se = 0.f, sw = 0.f;
#pragma unroll
    for (int r = 0; r < 16; ++r) {
      float e = __expf(vals[r] - m);
      se += e;
      sw += e * (float)r;
    }
    dist[f] = sw / se;
  }

  float* yr = ybuf + ((size_t)b * A_TOT + a) * 84;
  yr[0] = (ax - dist[0]) * s;
  yr[1] = (ay - dist[1]) * s;
  yr[2] = (ax + dist[2]) * s;
  yr[3] = (ay + dist[3]) * s;

  const float* sc = scoreAll + (size_t)b * NC_ * A_TOT + a;
  float mv = -__builtin_inff();
  for (int c = 0; c < NC_; ++c) {
    float v = 1.f / (1.f + __expf(-sc[(size_t)c * A_TOT]));
    yr[4 + c] = v;
    mv = fmaxf(mv, v);
  }
  maxsc[(size_t)b * A_TOT + a] = mv;
}

// ---------------------------------------------------------------------------
// Two-stage top-k (matches lax.top_k: descending, ties -> lowest index).
// One block per batch.
// ---------------------------------------------------------------------------
__global__ __launch_bounds__(256) void topk_kernel(
    const float* __restrict__ ybuf,   // (B,8400,84)
    const float* __restrict__ maxsc,  // (B,8400)
    float* __restrict__ sc2,          // (B, 300*80) scratch
    float* __restrict__ out)          // (B,300,6)
{
  const int b = blockIdx.x;
  const int tid = threadIdx.x;
  __shared__ float sv[A_TOT];
  __shared__ int   ori[MAX_DET_];
  __shared__ float rv[256];
  __shared__ int   ri[256];
  __shared__ unsigned int used[(MAX_DET_ * NC_ + 31) / 32];  // 750 words

  for (int i = tid; i < A_TOT; i += 256) sv[i] = maxsc[(size_t)b * A_TOT + i];
  for (int i = tid; i < (MAX_DET_ * NC_ + 31) / 32; i += 256) used[i] = 0u;
  __syncthreads();

  // phase 1: top-300 anchors by per-anchor max class score
  for (int k = 0; k < MAX_DET_; ++k) {
    float bv = -__builtin_inff(); int bi = 0x7fffffff;
    for (int i = tid; i < A_TOT; i += 256) {
      float v = sv[i];
      if (v > bv || (v == bv && i < bi)) { bv = v; bi = i; }
    }
    rv[tid] = bv; ri[tid] = bi;
    __syncthreads();
    for (int off = 128; off > 0; off >>= 1) {
      if (tid < off) {
        float v2 = rv[tid + off]; int i2 = ri[tid + off];
        if (v2 > rv[tid] || (v2 == rv[tid] && i2 < ri[tid])) { rv[tid] = v2; ri[tid] = i2; }
      }
      __syncthreads();
    }
    if (tid == 0) { ori[k] = ri[0]; sv[ri[0]] = -__builtin_inff(); }
    __syncthreads();
  }

  // phase 2: gather 300*80 candidate scores to global scratch
  float* scrow = sc2 + (size_t)b * (MAX_DET_ * NC_);
  for (int i = tid; i < MAX_DET_ * NC_; i += 256) {
    int j = i / NC_, c = i - j * NC_;
    scrow[i] = ybuf[((size_t)b * A_TOT + ori[j]) * 84 + 4 + c];
  }
  __threadfence();
  __syncthreads();

  // phase 3: top-300 of 24000 (exclusion via LDS bitmask; scrow stays immutable)
  for (int k = 0; k < MAX_DET_; ++k) {
    float bv = -__builtin_inff(); int bi = 0x7fffffff;
    for (int i = tid; i < MAX_DET_ * NC_; i += 256) {
      if ((used[i >> 5] >> (i & 31)) & 1u) continue;
      float v = scrow[i];
      if (v > bv || (v == bv && i < bi)) { bv = v; bi = i; }
    }
    rv[tid] = bv; ri[tid] = bi;
    __syncthreads();
    for (int off = 128; off > 0; off >>= 1) {
      if (tid < off) {
        float v2 = rv[tid + off]; int i2 = ri[tid + off];
        if (v2 > rv[tid] || (v2 == rv[tid] && i2 < ri[tid])) { rv[tid] = v2; ri[tid] = i2; }
      }
      __syncthreads();
    }
    if (tid == 0) {
      int idx = ri[0];
      used[idx >> 5] |= (1u << (idx & 31));
      int j = idx / NC_;
      int label = idx - j * NC_;
      int anchor = ori[j];
      const float* yr = ybuf + ((size_t)b * A_TOT + anchor) * 84;
      float* orow = out + ((size_t)b * MAX_DET_ + k) * 6;
      orow[0] = yr[0]; orow[1] = yr[1]; orow[2] = yr[2]; orow[3] = yr[3];
      orow[4] = rv[0]; orow[5] = (float)label;
    }
    __syncthreads();
  }
}

// ---------------------------------------------------------------------------
extern "C" void kernel_launch(void* const* d_in, const int* in_sizes, int n_in,
                              void* d_out, int out_size, void* d_ws, size_t ws_size,
                              hipStream_t stream)
{
  (void)in_sizes; (void)n_in; (void)out_size; (void)ws_size;

  // setup_inputs() order: x0,x1,x2, then cv2[0..2]{w1,b1,w2,b2,w3,b3},
  // then cv3[0..2]{dw1,db1,w1,b1,dw2,db2,w2,b2,w3,b3}  (51 inputs)
  const float* X[3] = {(const float*)d_in[0], (const float*)d_in[1], (const float*)d_in[2]};
  const float* c2[3][6];
  const float* c3[3][10];
  int p = 3;
  for (int i = 0; i < 3; ++i)
    for (int j = 0; j < 6; ++j) c2[i][j] = (const float*)d_in[p++];
  for (int i = 0; i < 3; ++i)
    for (int j = 0; j < 10; ++j) c3[i][j] = (const float*)d_in[p++];

  float* ws = (float*)d_ws;
  const size_t BUF = (size_t)BATCH * 256 * 6400;  // 13,107,200 floats
  float* bufA     = ws;
  float* bufB     = ws + BUF;
  float* boxAll   = ws + 2 * BUF;                               // (8,64,8400)
  float* scoreAll = boxAll + (size_t)BATCH * 64 * A_TOT;        // (8,80,8400)
  // after convs finish, bufA/bufB are free -> alias decode/top-k buffers
  float* ybuf  = bufA;                                          // (8,8400,84)
  float* maxsc = bufB;                                          // (8,8400)
  float* sc2   = bufB + (size_t)BATCH * A_TOT;                  // (8,24000)

  const int CHs[3]   = {256, 512, 1024};
  const int Hs[3]    = {80, 40, 20};
  const int aoffs[3] = {0, 6400, 8000};

  for (int i = 0; i < 3; ++i) {
    const int c = CHs[i], H = Hs[i], W = Hs[i], HW = H * W;
    const dim3 blk(256);
    const int nPix = (HW + 63) / 64;

    // --- cv2 (box branch) ---
    conv_wmma<<<dim3(nPix, 1, BATCH), blk, 0, stream>>>(
        X[i], c2[i][0], c2[i][1], bufA, c, 64, H, W, 1, 1, HW, 0);
    conv_wmma<<<dim3(nPix, 1, BATCH), blk, 0, stream>>>(
        bufA, c2[i][2], c2[i][3], bufB, 64, 64, H, W, 1, 1, HW, 0);
    conv_wmma<<<dim3(nPix, 1, BATCH), blk, 0, stream>>>(
        bufB, c2[i][4], c2[i][5], boxAll, 64, 64, H, W, 0, 0, A_TOT, aoffs[i]);

    // --- cv3 (cls branch) ---
    int tot1 = BATCH * c * HW;
    dwconv3x3_silu<<<(tot1 + 255) / 256, 256, 0, stream>>>(
        X[i], c3[i][0], c3[i][1], bufA, c, H, W, tot1);
    conv_wmma<<<dim3(nPix, 4, BATCH), blk, 0, stream>>>(
        bufA, c3[i][2], c3[i][3], bufB, c, 256, H, W, 0, 1, HW, 0);
    int tot2 = BATCH * 256 * HW;
    dwconv3x3_silu<<<(tot2 + 255) / 256, 256, 0, stream>>>(
        bufB, c3[i][4], c3[i][5], bufA, 256, H, W, tot2);
    conv_wmma<<<dim3(nPix, 4, BATCH), blk, 0, stream>>>(
        bufA, c3[i][6], c3[i][7], bufB, 256, 256, H, W, 0, 1, HW, 0);
    conv_wmma<<<dim3(nPix, 2, BATCH), blk, 0, stream>>>(
        bufB, c3[i][8], c3[i][9], scoreAll, 256, 80, H, W, 0, 0, A_TOT, aoffs[i]);
  }

  int totD = BATCH * A_TOT;
  decode_kernel<<<(totD + 255) / 256, 256, 0, stream>>>(boxAll, scoreAll, ybuf, maxsc);
  topk_kernel<<<BATCH, 256, 0, stream>>>(ybuf, maxsc, sc2, (float*)d_out);
}